// LockAndKeyBridge_3530463117431
// MI455X (gfx1250) — compile-verified
//
#include <hip/hip_runtime.h>
#include <hip/hip_bf16.h>
#include <math.h>
#include <stdint.h>

#define B_ 64
#define K_ 64
#define D_ 256
#define R_ 128
#define BK (B_ * K_)          // 4096 rows
#define NTOK (BK * D_)        // 1048576 soft-token elements
#define TOPK_IDX 11           // k = int(128*0.1) = 12  -> threshold = rank 11 (0-based)

#define HR_STRIDE 66                       // floats; 264B rows: 8B-aligned, conflict-free reads
#define HR_BUF_FLOATS (R_ * HR_STRIDE)     // 8448 floats = 33792 bytes per buffer
#define HR_BUF_BYTES (HR_BUF_FLOATS * 4)

typedef float v2f __attribute__((ext_vector_type(2)));
typedef float v8f __attribute__((ext_vector_type(8)));

// ---------------------------------------------------------------------------
// Kernel 1: f32 WMMA GEMM.
//   hk[4096,256] = keys[4096,256] @ W1[0:256, :]     (W1 is row-major (in,out))
//   hr[ 128,256] = receptors[128,256] @ W1[256:512, :]
// One wave (32 threads) per 16x16 output tile; K-loop of 64 x wmma_f32_16x16x4_f32.
// ---------------------------------------------------------------------------
__global__ __launch_bounds__(32)
void lk_gemm_wmma(const float* __restrict__ keys,
                  const float* __restrict__ receptors,
                  const float* __restrict__ W1,
                  float* __restrict__ hk,
                  float* __restrict__ hr)
{
    const int tile    = blockIdx.x;      // 0 .. 264*16-1
    const int rowTile = tile >> 4;       // 0..263  (0..255 -> keys, 256..263 -> receptors)
    const int colTile = tile & 15;
    const int col0    = colTile * 16;

    const float* A;
    const float* W;
    float*       Cout;
    int          arow0;
    if (rowTile < 256) {
        A     = keys;
        W     = W1;                      // top half of W1
        Cout  = hk;
        arow0 = rowTile * 16;
    } else {
        A     = receptors;
        W     = W1 + D_ * D_;            // bottom half of W1
        Cout  = hr;
        arow0 = (rowTile - 256) * 16;
    }

    const int lane = threadIdx.x;        // wave32
    const int m    = lane & 15;          // M index (A rows / D cols)
    const int kh   = (lane >> 4) * 2;    // K sub-offset: 0 for lanes 0-15, 2 for 16-31

    v8f c = {};
    const float* arow = A + (arow0 + m) * D_;   // this lane's A row
    #pragma unroll 4
    for (int k = 0; k < D_; k += 4) {
        v2f a, b;
        // A 16x4 tile: VGPR j holds K = k + kh + j, M = lane%16
        a.x = arow[k + kh + 0];
        a.y = arow[k + kh + 1];
        // B 4x16 tile: VGPR j holds row K = k + kh + j, N = lane%16
        b.x = W[(k + kh + 0) * D_ + col0 + m];
        b.y = W[(k + kh + 1) * D_ + col0 + m];
        c = __builtin_amdgcn_wmma_f32_16x16x4_f32(
                /*neg_a=*/false, a, /*neg_b=*/false, b,
                /*c_mod=*/(short)0, c, /*reuse_a=*/false, /*reuse_b=*/false);
    }

    // C/D layout: VGPR v -> rows M=v (lanes 0-15) and M=v+8 (lanes 16-31), N = lane%16
    const int rbase = arow0 + 8 * (lane >> 4);
    #pragma unroll
    for (int v = 0; v < 8; ++v) {
        Cout[(rbase + v) * D_ + col0 + m] = c[v];
    }
}

// ---------------------------------------------------------------------------
// Block-wide reductions for blockDim.x == 128
// ---------------------------------------------------------------------------
__device__ __forceinline__ float blockSum128(float v, float* red, int t) {
    red[t] = v; __syncthreads();
    #pragma unroll
    for (int s = 64; s > 0; s >>= 1) {
        if (t < s) red[t] += red[t + s];
        __syncthreads();
    }
    float r = red[0]; __syncthreads();
    return r;
}

__device__ __forceinline__ float blockMax128(float v, float* red, int t) {
    red[t] = v; __syncthreads();
    #pragma unroll
    for (int s = 64; s > 0; s >>= 1) {
        if (t < s) red[t] = fmaxf(red[t], red[t + s]);
        __syncthreads();
    }
    float r = red[0]; __syncthreads();
    return r;
}

// ---------------------------------------------------------------------------
// Kernel 2: fused per-(b,k) pipeline. One block of 128 threads per bk row.
// hr is staged into LDS via double-buffered GLOBAL_LOAD_ASYNC_TO_LDS_B64
// (ASYNCcnt), overlapping the next chunk's DMA with the current gelu chunk.
// ---------------------------------------------------------------------------
__global__ __launch_bounds__(128)
void lk_bind_fused(const float* __restrict__ hk,
                   const float* __restrict__ hr,
                   const float* __restrict__ keys,
                   const float* __restrict__ receptors,
                   const float* __restrict__ b1,
                   const float* __restrict__ W2,
                   const float* __restrict__ b2,
                   const float* __restrict__ out_scale,
                   float* __restrict__ soft_tokens,
                   float* __restrict__ row_sq,
                   float* __restrict__ row_ent,
                   float* __restrict__ row_ts,
                   float* __restrict__ row_tq)
{
    __shared__ float hr_s[2 * HR_BUF_FLOATS];  // double-buffered 64-wide d-chunks of hr
    __shared__ float hkb_s[D_];                // hk row + b1
    __shared__ float w2_s[D_];
    __shared__ float aff_s[R_];
    __shared__ float bind_s[R_];
    __shared__ float red_s[R_];
    __shared__ float thr_s;

    const int bk = blockIdx.x;        // 0..4095
    const int t  = threadIdx.x;       // 0..127

    // stage hk row (+b1) and W2
    hkb_s[t]        = hk[bk * D_ + t]        + b1[t];
    hkb_s[t + 128]  = hk[bk * D_ + t + 128]  + b1[t + 128];
    w2_s[t]         = W2[t];
    w2_s[t + 128]   = W2[t + 128];

    const uint32_t hr_lds_base = (uint32_t)(uintptr_t)(&hr_s[0]);
    const uint64_t hr_gbase    = (uint64_t)(uintptr_t)hr;

    // issue async DMA of hr[:, c*64 : (c+1)*64] into LDS buffer `buf`
    auto issue_chunk = [&](int c, int buf) {
        const uint32_t lbase = hr_lds_base + (uint32_t)buf * (uint32_t)HR_BUF_BYTES;
        #pragma unroll 4
        for (int i = 0; i < 32; ++i) {
            int idx = i * 128 + t;        // 0..4095 B64 transfers for this chunk
            int r   = idx >> 5;           // receptor row
            int d2  = idx & 31;           // float-pair within chunk
            uint32_t goff  = (uint32_t)(r * (D_ * 4) + c * 256 + d2 * 8);
            uint32_t laddr = lbase + (uint32_t)(r * (HR_STRIDE * 4) + d2 * 8);
            asm volatile("global_load_async_to_lds_b64 %0, %1, %2"
                         :: "v"(laddr), "v"(goff), "s"(hr_gbase)
                         : "memory");
        }
    };

    issue_chunk(0, 0);

    // ---- affinity[r] = sum_d gelu(hkb + hr) * W2  (d chunked by 64 through LDS) ----
    float acc = 0.0f;
    #pragma unroll 1
    for (int c = 0; c < 4; ++c) {
        const int buf = c & 1;
        // wait for this wave's outstanding async DMA (chunk c), then make all
        // waves' LDS deposits visible block-wide.
        asm volatile("s_wait_asynccnt 0x0" ::: "memory");
        __syncthreads();
        // overlap: start DMA of chunk c+1 into the other buffer
        if (c < 3) issue_chunk(c + 1, buf ^ 1);

        const float* hrow = &hr_s[buf * HR_BUF_FLOATS + t * HR_STRIDE];
        const float* hb   = &hkb_s[c * 64];
        const float* w2c  = &w2_s[c * 64];
        #pragma unroll 4
        for (int d = 0; d < 64; ++d) {
            float x = hb[d] + hrow[d];
            float g = 0.5f * x * (1.0f + erff(x * 0.70710678118654752f));
            acc = fmaf(g, w2c[d], acc);
        }
    }
    float a = acc + b2[0];
    aff_s[t] = a;
    __syncthreads();

    // ---- top-k threshold: value with exactly 11 strictly-greater (tie-broken) ----
    {
        int cnt = 0;
        #pragma unroll 4
        for (int j = 0; j < R_; ++j) {
            float aj = aff_s[j];
            cnt += (aj > a) || (aj == a && j < t);
        }
        if (cnt == TOPK_IDX) thr_s = a;
    }
    __syncthreads();
    const float thr = thr_s;

    // ---- softmax * sigmoid mask ----
    float mx = blockMax128(a, red_s, t);
    float e  = __expf(a - mx);
    float se = blockSum128(e, red_s, t);
    float soft = e / se;
    float mask = 1.0f / (1.0f + __expf(-10.0f * (a - thr)));
    float bind = soft * mask;
    bind_s[t] = bind;

    // per-row sparsity / entropy partials
    float sq  = blockSum128(bind * bind, red_s, t);
    float ent = blockSum128(bind * __logf(bind + 1e-8f), red_s, t);
    if (t == 0) { row_sq[bk] = sq; row_ent[bk] = ent; }
    __syncthreads();

    // ---- bound = binding @ receptors ; soft tokens; RMS norm ----
    const int d0 = t, d1 = t + 128;
    float acc0 = 0.0f, acc1 = 0.0f;
    #pragma unroll 4
    for (int r = 0; r < R_; ++r) {
        float br = bind_s[r];
        acc0 = fmaf(br, receptors[r * D_ + d0], acc0);
        acc1 = fmaf(br, receptors[r * D_ + d1], acc1);
    }
    float st0 = keys[bk * D_ + d0] + acc0;
    float st1 = keys[bk * D_ + d1] + acc1;

    float ssum = blockSum128(st0 * st0 + st1 * st1, red_s, t);
    float rms  = sqrtf(ssum * (1.0f / (float)D_) + 1e-8f);
    float scl  = out_scale[0] / rms;
    float o0 = st0 * scl, o1 = st1 * scl;
    soft_tokens[bk * D_ + d0] = o0;
    soft_tokens[bk * D_ + d1] = o1;

    // variance partials
    float ts = blockSum128(o0 + o1, red_s, t);
    float tq = blockSum128(o0 * o0 + o1 * o1, red_s, t);
    if (t == 0) { row_ts[bk] = ts; row_tq[bk] = tq; }
}

// ---------------------------------------------------------------------------
// Kernel 3: reduce 4096-row partials into the 3 scalar outputs.
// ---------------------------------------------------------------------------
__global__ __launch_bounds__(256)
void lk_finalize(const float* __restrict__ row_sq,
                 const float* __restrict__ row_ent,
                 const float* __restrict__ row_ts,
                 const float* __restrict__ row_tq,
                 float* __restrict__ scalars)   // [sparsity_loss, entropy, token_var]
{
    __shared__ float s0[256], s1[256], s2[256], s3[256];
    const int t = threadIdx.x;
    float a0 = 0.f, a1 = 0.f, a2 = 0.f, a3 = 0.f;
    for (int i = t; i < BK; i += 256) {
        a0 += row_sq[i];
        a1 += row_ent[i];
        a2 += row_ts[i];
        a3 += row_tq[i];
    }
    s0[t] = a0; s1[t] = a1; s2[t] = a2; s3[t] = a3;
    __syncthreads();
    for (int s = 128; s > 0; s >>= 1) {
        if (t < s) {
            s0[t] += s0[t + s]; s1[t] += s1[t + s];
            s2[t] += s2[t + s]; s3[t] += s3[t + s];
        }
        __syncthreads();
    }
    if (t == 0) {
        const float invBK = 1.0f / (float)BK;
        scalars[0] = s0[0] * invBK;                 // sparsity_loss
        scalars[1] = -s1[0] * invBK;                // binding_entropy
        const double n    = (double)NTOK;
        const double sum  = (double)s2[0];
        const double sumq = (double)s3[0];
        scalars[2] = (float)((sumq - sum * sum / n) / (n - 1.0));  // unbiased var
    }
}

// ---------------------------------------------------------------------------
extern "C" void kernel_launch(void* const* d_in, const int* in_sizes, int n_in,
                              void* d_out, int out_size, void* d_ws, size_t ws_size,
                              hipStream_t stream) {
    const float* keys      = (const float*)d_in[0];  // [64,64,256]
    const float* receptors = (const float*)d_in[1];  // [128,256]
    const float* W1        = (const float*)d_in[2];  // [512,256]
    const float* b1        = (const float*)d_in[3];  // [256]
    const float* W2        = (const float*)d_in[4];  // [256]
    const float* b2        = (const float*)d_in[5];  // scalar
    const float* oscale    = (const float*)d_in[6];  // scalar

    float* out = (float*)d_out;                      // [1048576 soft_tokens][3 scalars]

    float* ws      = (float*)d_ws;
    float* hk      = ws;                             // 4096*256
    float* hr      = hk + (size_t)BK * D_;           // 128*256
    float* row_sq  = hr + (size_t)R_ * D_;           // 4096
    float* row_ent = row_sq + BK;                    // 4096
    float* row_ts  = row_ent + BK;                   // 4096
    float* row_tq  = row_ts + BK;                    // 4096

    // 1) f32 WMMA GEMM: hk and hr. (256 + 8) row-tiles x 16 col-tiles, 1 wave each.
    lk_gemm_wmma<<<dim3(264 * 16), dim3(32), 0, stream>>>(keys, receptors, W1, hk, hr);

    // 2) fused affinity / topk / softmax / binding / bound / rms-norm
    lk_bind_fused<<<dim3(BK), dim3(128), 0, stream>>>(hk, hr, keys, receptors,
                                                      b1, W2, b2, oscale,
                                                      out, row_sq, row_ent, row_ts, row_tq);

    // 3) scalar reductions
    lk_finalize<<<dim3(1), dim3(256), 0, stream>>>(row_sq, row_ent, row_ts, row_tq,
                                                   out + (size_t)NTOK);
}